// StaticGNN_Identity_53171695124550
// MI455X (gfx1250) — compile-verified
//
#include <hip/hip_runtime.h>
#include <hip/hip_bf16.h>

typedef __attribute__((ext_vector_type(2))) float v2f;
typedef __attribute__((ext_vector_type(8))) float v8f;

// ---------- helpers ----------
__device__ __forceinline__ unsigned f2ord(float f) {
  unsigned u = __float_as_uint(f);
  return (u & 0x80000000u) ? ~u : (u | 0x80000000u);
}
__device__ __forceinline__ float ord2f(unsigned u) {
  return (u & 0x80000000u) ? __uint_as_float(u & 0x7FFFFFFFu) : __uint_as_float(~u);
}

static inline int cdiv(long long a, long long b) { return (int)((a + b - 1) / b); }

// ---------- WMMA f32 16x16x4 core: one wave computes 16 rows x 64 cols ----------
// Software-pipelined: next k-step A/B fragments are loaded before issuing the
// current step's 4 WMMAs, so the wait in front of the WMMA group covers loads
// issued one iteration earlier. K and LDW are compile-time for immediate-offset
// addressing and full unroll.
// A frag (16x4):  lane<16 -> {K0,K1}, lane>=16 -> {K2,K3}, row = lane&15
// B frag (4x16):  lane<16 -> {K0,K1}, lane>=16 -> {K2,K3}, col = lane&15
// C/D (16x16):    v8f elem i -> row = i + 8*(lane>=16), col = lane&15
template <int K, int LDW>
__device__ __forceinline__ void wmma_tile_16x64_t(const float* __restrict__ ar,
                                                  const float* __restrict__ Wm,
                                                  int col0, v8f acc[4]) {
  const int lane = threadIdx.x & 31, half = lane >> 4, l16 = lane & 15;
  const float* abase = ar + half * 2;
  const float* wbase = Wm + col0 + l16 + (size_t)(half * 2) * LDW;
  v2f a_cur; a_cur.x = abase[0]; a_cur.y = abase[1];
  v2f b_cur[4];
#pragma unroll
  for (int t = 0; t < 4; ++t) { b_cur[t].x = wbase[t * 16]; b_cur[t].y = wbase[t * 16 + LDW]; }
#pragma unroll
  for (int k = 0; k < K; k += 4) {
    const int kn = (k + 4 < K) ? (k + 4) : 0;       // clamp: avoid OOB/undef on last iter
    v2f a_nxt; a_nxt.x = abase[kn]; a_nxt.y = abase[kn + 1];
    v2f b_nxt[4];
    const float* wn = wbase + (size_t)kn * LDW;
#pragma unroll
    for (int t = 0; t < 4; ++t) { b_nxt[t].x = wn[t * 16]; b_nxt[t].y = wn[t * 16 + LDW]; }
#pragma unroll
    for (int t = 0; t < 4; ++t)
      acc[t] = __builtin_amdgcn_wmma_f32_16x16x4_f32(false, a_cur, false, b_cur[t],
                                                     (short)0, acc[t], false, false);
    a_cur = a_nxt;
#pragma unroll
    for (int t = 0; t < 4; ++t) b_cur[t] = b_nxt[t];
  }
}

// ---------- plain GEMM: C = A @ W  (A:[M,K], W:[K,NC], C:[M,NC]) ----------
template <int K, int NC>
__global__ void k_gemm_t(const float* __restrict__ A, const float* __restrict__ Wm,
                         float* __restrict__ C, int M) {
  const int row0 = blockIdx.x * 16, col0 = blockIdx.y * 64;
  const int lane = threadIdx.x & 31, half = lane >> 4, l16 = lane & 15;
  int arow = row0 + l16; if (arow >= M) arow = M - 1;
  v8f acc[4] = { {0,0,0,0,0,0,0,0}, {0,0,0,0,0,0,0,0}, {0,0,0,0,0,0,0,0}, {0,0,0,0,0,0,0,0} };
  wmma_tile_16x64_t<K, NC>(A + (size_t)arow * K, Wm, col0, acc);
#pragma unroll
  for (int i = 0; i < 8; ++i) {
    const int row = row0 + i + half * 8;
    if (row < M) {
#pragma unroll
      for (int t = 0; t < 4; ++t) C[(size_t)row * NC + col0 + t * 16 + l16] = acc[t][i];
    }
  }
}

// ---------- edge_proj: relu(edge_attr@ep_w + b), scatter-add to x_out|x_in ----------
__global__ void k_edge_proj_scatter(const float* __restrict__ EA, const float* __restrict__ Wm,
                                    const float* __restrict__ bias,
                                    const int* __restrict__ src, const int* __restrict__ dst,
                                    float* __restrict__ xin, int E) {
  const int row0 = blockIdx.x * 16, col0 = blockIdx.y * 64;
  const int lane = threadIdx.x & 31, half = lane >> 4, l16 = lane & 15;
  int arow = row0 + l16; if (arow >= E) arow = E - 1;
  v8f acc[4] = { {0,0,0,0,0,0,0,0}, {0,0,0,0,0,0,0,0}, {0,0,0,0,0,0,0,0}, {0,0,0,0,0,0,0,0} };
  wmma_tile_16x64_t<64, 128>(EA + (size_t)arow * 64, Wm, col0, acc);
#pragma unroll
  for (int i = 0; i < 8; ++i) {
    const int row = row0 + i + half * 8;
    if (row >= E) continue;
    const int s = src[row], d = dst[row];
#pragma unroll
    for (int t = 0; t < 4; ++t) {
      const int col = col0 + t * 16 + l16;
      float v = acc[t][i] + bias[col];
      v = fmaxf(v, 0.f);
      atomicAdd(&xin[(size_t)s * 256 + col], v);
      atomicAdd(&xin[(size_t)d * 256 + 128 + col], v);
    }
  }
}

// ---------- degree counts + self-loop attr accumulation ----------
__global__ void k_counts(const int* __restrict__ src, const int* __restrict__ dst, int E,
                         float* cnt_out, float* cnt_in) {
  int e = blockIdx.x * blockDim.x + threadIdx.x;
  if (e >= E) return;
  atomicAdd(&cnt_out[src[e]], 1.f);
  atomicAdd(&cnt_in[dst[e]], 1.f);
}
__global__ void k_loop_scatter(const float* __restrict__ ea, const int* __restrict__ dst,
                               int E, float* loopa) {
  int g = blockIdx.x * blockDim.x + threadIdx.x;
  if (g >= E * 64) return;
  int e = g >> 6, c = g & 63;
  atomicAdd(&loopa[(size_t)dst[e] * 64 + c], ea[g]);
}
__global__ void k_norm_xinput(float* xin, const float* co, const float* ci, int N) {
  int g = blockIdx.x * blockDim.x + threadIdx.x;
  if (g >= N * 256) return;
  int n = g >> 8, c = g & 255;
  xin[g] /= fmaxf(c < 128 ? co[n] : ci[n], 1.f);
}
__global__ void k_norm_loopa(float* loopa, const float* ci, int N) {
  int g = blockIdx.x * blockDim.x + threadIdx.x;
  if (g >= N * 64) return;
  loopa[g] /= fmaxf(ci[g >> 6], 1.f);
}
__global__ void k_init_amax(unsigned* amax, int n) {
  int g = blockIdx.x * blockDim.x + threadIdx.x;
  if (g < n) amax[g] = 0x007FFFFFu;  // f2ord(-inf)
}

// ---------- GATv2 pass1: fused edge transform + leaky_relu + att dot + seg max ----------
__global__ void k_gat_pass1(const float* __restrict__ xl, const float* __restrict__ xr,
                            const float* __restrict__ we, const float* __restrict__ att,
                            const float* __restrict__ edge_attr, const float* __restrict__ loopa,
                            const int* __restrict__ src, const int* __restrict__ dst,
                            int E, float* __restrict__ logits, unsigned* __restrict__ amax, int H) {
  const int e = blockIdx.x, t = threadIdx.x;  // 128 threads
  __shared__ float ea[64];
  __shared__ float red[256];
  int s_, d_;
  const float* arow;
  if (e < E) { s_ = src[e]; d_ = dst[e]; arow = edge_attr + (size_t)e * 64; }
  else       { s_ = d_ = e - E;          arow = loopa     + (size_t)(e - E) * 64; }
  if (t < 64) ea[t] = arow[t];
  __syncthreads();
  const int HC = H * 128;
  for (int h = 0; h < H; ++h) {
    const int ch = h * 128 + t;
    float acc = 0.f;
#pragma unroll 8
    for (int k = 0; k < 64; ++k) acc += ea[k] * we[k * HC + ch];
    float m = xl[(size_t)s_ * HC + ch] + xr[(size_t)d_ * HC + ch] + acc;
    m = m > 0.f ? m : 0.2f * m;                 // leaky_relu(0.2)
    red[h * 128 + t] = m * att[h * 128 + t];
  }
  __syncthreads();
  for (int st = 64; st > 0; st >>= 1) {
    if (t < st) { red[t] += red[t + st]; if (H == 2) red[128 + t] += red[128 + t + st]; }
    __syncthreads();
  }
  if (t < H) {
    const float lg = red[t * 128];
    logits[(size_t)e * H + t] = lg;
    atomicMax(&amax[(size_t)d_ * H + t], f2ord(lg));
  }
}

// ---------- pass2: exp(logit - max) and segment denominator ----------
__global__ void k_gat_pass2(float* __restrict__ logits, const unsigned* __restrict__ amax,
                            float* __restrict__ denom, const int* __restrict__ dst,
                            int E, int Ea, int H) {
  int g = blockIdx.x * blockDim.x + threadIdx.x;
  if (g >= Ea * H) return;
  const int e = g / H, h = g % H;
  const int d_ = (e < E) ? dst[e] : e - E;
  float m = ord2f(amax[(size_t)d_ * H + h]);
  if (!(m > -1e38f && m < 1e38f)) m = 0.f;      // robust finiteness check
  const float ev = expf(logits[g] - m);
  logits[g] = ev;
  atomicAdd(&denom[(size_t)d_ * H + h], ev);
}

// ---------- pass3: out[dst] += xl[src] * alpha ----------
__global__ void k_gat_pass3(const float* __restrict__ xl, const float* __restrict__ lexp,
                            const float* __restrict__ denom,
                            const int* __restrict__ src, const int* __restrict__ dst,
                            int E, int Ea, int H, float* __restrict__ hacc) {
  const size_t g = (size_t)blockIdx.x * blockDim.x + threadIdx.x;
  const int HC = H * 128;
  if (g >= (size_t)Ea * HC) return;
  const int e = (int)(g / HC), ch = (int)(g % HC), h = ch >> 7;
  const int s_ = (e < E) ? src[e] : e - E;
  const int d_ = (e < E) ? dst[e] : e - E;
  const float alpha = lexp[(size_t)e * H + h] / (denom[(size_t)d_ * H + h] + 1e-16f);
  atomicAdd(&hacc[(size_t)d_ * HC + ch], xl[(size_t)s_ * HC + ch] * alpha);
}

// ---------- head-mean + bias + layernorm + elu ----------
__global__ void k_gat_finalize(const float* __restrict__ hacc, const float* __restrict__ bias,
                               const float* __restrict__ lng, const float* __restrict__ lnb,
                               int H, float* __restrict__ hout) {
  const int n = blockIdx.x, t = threadIdx.x;  // 128 threads
  float v = (H == 2) ? 0.5f * (hacc[(size_t)n * 256 + t] + hacc[(size_t)n * 256 + 128 + t]) + bias[t]
                     : hacc[(size_t)n * 128 + t] + bias[t];
  __shared__ float red[128];
  red[t] = v; __syncthreads();
  for (int s = 64; s > 0; s >>= 1) { if (t < s) red[t] += red[t + s]; __syncthreads(); }
  const float mu = red[0] * (1.f / 128.f);
  __syncthreads();
  const float dv = v - mu;
  red[t] = dv * dv; __syncthreads();
  for (int s = 64; s > 0; s >>= 1) { if (t < s) red[t] += red[t + s]; __syncthreads(); }
  const float var = red[0] * (1.f / 128.f);
  float y = dv * rsqrtf(var + 1e-5f) * lng[t] + lnb[t];
  hout[(size_t)n * 128 + t] = y > 0.f ? y : (expf(y) - 1.f);
}

// ---------- fused WMMA classifier: [h2[s]|h2[d]|eattr](320) -> 128 -> 64 -> 1 ----------
#define RLD 321
#define Z1LD 129
#define Z2LD 65
__global__ void k_classifier(const float* __restrict__ h2, const float* __restrict__ edge_attr,
                             const int* __restrict__ src, const int* __restrict__ dst,
                             const float* __restrict__ c1w, const float* __restrict__ c1b,
                             const float* __restrict__ c2w, const float* __restrict__ c2b,
                             const float* __restrict__ c3w, const float* __restrict__ c3b,
                             float* __restrict__ out, int E) {
  __shared__ float rep[16 * RLD];
  __shared__ float z1[16 * Z1LD];
  __shared__ float z2[16 * Z2LD];
  const int e0 = blockIdx.x * 16, tid = threadIdx.x;  // 128 threads = 4 waves
  for (int idx = tid; idx < 16 * 320; idx += 128) {
    const int r = idx / 320, c = idx % 320;
    int e = e0 + r; if (e >= E) e = E - 1;
    float v;
    if (c < 128)       v = h2[(size_t)src[e] * 128 + c];
    else if (c < 256)  v = h2[(size_t)dst[e] * 128 + (c - 128)];
    else               v = edge_attr[(size_t)e * 64 + (c - 256)];
    rep[r * RLD + c] = v;
  }
  __syncthreads();
  const int wave = tid >> 5, lane = tid & 31, half = lane >> 4, l16 = lane & 15;
  {   // z1 = relu(rep @ c1w + c1b): each wave -> 32 cols, pipelined K=320
    v8f acc[2] = { {0,0,0,0,0,0,0,0}, {0,0,0,0,0,0,0,0} };
    const int colbase = wave * 32;
    const float* abase = rep + l16 * RLD + half * 2;
    const float* wbase = c1w + colbase + l16 + (size_t)(half * 2) * 128;
    v2f a_cur; a_cur.x = abase[0]; a_cur.y = abase[1];
    v2f b_cur[2];
#pragma unroll
    for (int t = 0; t < 2; ++t) { b_cur[t].x = wbase[t * 16]; b_cur[t].y = wbase[t * 16 + 128]; }
#pragma unroll
    for (int k = 0; k < 320; k += 4) {
      const int kn = (k + 4 < 320) ? (k + 4) : 0;
      v2f a_nxt; a_nxt.x = abase[kn]; a_nxt.y = abase[kn + 1];
      v2f b_nxt[2];
      const float* wn = wbase + (size_t)kn * 128;
#pragma unroll
      for (int t = 0; t < 2; ++t) { b_nxt[t].x = wn[t * 16]; b_nxt[t].y = wn[t * 16 + 128]; }
#pragma unroll
      for (int t = 0; t < 2; ++t)
        acc[t] = __builtin_amdgcn_wmma_f32_16x16x4_f32(false, a_cur, false, b_cur[t],
                                                       (short)0, acc[t], false, false);
      a_cur = a_nxt;
#pragma unroll
      for (int t = 0; t < 2; ++t) b_cur[t] = b_nxt[t];
    }
#pragma unroll
    for (int t = 0; t < 2; ++t)
#pragma unroll
      for (int i = 0; i < 8; ++i) {
        const int row = i + half * 8, col = colbase + t * 16 + l16;
        const float v = acc[t][i] + c1b[col];
        z1[row * Z1LD + col] = fmaxf(v, 0.f);
      }
  }
  __syncthreads();
  {   // z2 = relu(z1 @ c2w + c2b): each wave -> 16 cols, pipelined K=128
    v8f acc = {0,0,0,0,0,0,0,0};
    const int colbase = wave * 16;
    const float* abase = z1 + l16 * Z1LD + half * 2;
    const float* wbase = c2w + colbase + l16 + (size_t)(half * 2) * 64;
    v2f a_cur; a_cur.x = abase[0]; a_cur.y = abase[1];
    v2f b_cur;  b_cur.x = wbase[0]; b_cur.y = wbase[64];
#pragma unroll
    for (int k = 0; k < 128; k += 4) {
      const int kn = (k + 4 < 128) ? (k + 4) : 0;
      v2f a_nxt; a_nxt.x = abase[kn]; a_nxt.y = abase[kn + 1];
      v2f b_nxt; b_nxt.x = wbase[(size_t)kn * 64]; b_nxt.y = wbase[(size_t)kn * 64 + 64];
      acc = __builtin_amdgcn_wmma_f32_16x16x4_f32(false, a_cur, false, b_cur,
                                                  (short)0, acc, false, false);
      a_cur = a_nxt; b_cur = b_nxt;
    }
#pragma unroll
    for (int i = 0; i < 8; ++i) {
      const int row = i + half * 8, col = colbase + l16;
      const float v = acc[i] + c2b[col];
      z2[row * Z2LD + col] = fmaxf(v, 0.f);
    }
  }
  __syncthreads();
  if (tid < 16) {   // out = z2 @ c3w + c3b
    float s = 0.f;
#pragma unroll 8
    for (int k = 0; k < 64; ++k) s += z2[tid * Z2LD + k] * c3w[k];
    if (e0 + tid < E) out[e0 + tid] = s + c3b[0];
  }
}

// ---------- host orchestration ----------
extern "C" void kernel_launch(void* const* d_in, const int* in_sizes, int n_in,
                              void* d_out, int out_size, void* d_ws, size_t ws_size,
                              hipStream_t stream) {
  const int*   eidx      = (const int*)d_in[1];
  const float* edge_attr = (const float*)d_in[2];
  const float* ep_w  = (const float*)d_in[3];
  const float* ep_b  = (const float*)d_in[4];
  const float* g1_wl = (const float*)d_in[5];
  const float* g1_wr = (const float*)d_in[6];
  const float* g1_we = (const float*)d_in[7];
  const float* g1_att= (const float*)d_in[8];
  const float* g1_b  = (const float*)d_in[9];
  const float* n1_g  = (const float*)d_in[10];
  const float* n1_b  = (const float*)d_in[11];
  const float* g2_wl = (const float*)d_in[12];
  const float* g2_wr = (const float*)d_in[13];
  const float* g2_we = (const float*)d_in[14];
  const float* g2_att= (const float*)d_in[15];
  const float* g2_b  = (const float*)d_in[16];
  const float* n2_g  = (const float*)d_in[17];
  const float* n2_b  = (const float*)d_in[18];
  const float* c1_w  = (const float*)d_in[19];
  const float* c1_b  = (const float*)d_in[20];
  const float* c2_w  = (const float*)d_in[21];
  const float* c2_b  = (const float*)d_in[22];
  const float* c3_w  = (const float*)d_in[23];
  const float* c3_b  = (const float*)d_in[24];

  const int N  = in_sizes[0] / 128;
  const int E  = in_sizes[1] / 2;
  const int Ea = E + N;
  const int* srcp = eidx;
  const int* dstp = eidx + E;

  float* Wf   = (float*)d_ws;
  float* XIN  = Wf;                                // N*256 (reused as HACC)
  float* CNTO = XIN  + (size_t)N * 256;            // N
  float* CNTI = CNTO + N;                          // N
  float* LOOPA= CNTI + N;                          // N*64
  float* XL   = LOOPA + (size_t)N * 64;            // N*256
  float* XR   = XL   + (size_t)N * 256;            // N*256
  float* LOG  = XR   + (size_t)N * 256;            // Ea*2
  unsigned* AMAX = (unsigned*)(LOG + (size_t)Ea * 2);  // N*2
  float* DEN  = (float*)AMAX + (size_t)N * 2;      // N*2
  float* H1   = DEN  + (size_t)N * 2;              // N*128
  float* H2   = H1   + (size_t)N * 128;            // N*128
  float* HACC = XIN;
  float* outp = (float*)d_out;

  // --- edge aggregation into x_input ---
  hipMemsetAsync(XIN, 0, (size_t)N * 322 * sizeof(float), stream);  // XIN + counts + LOOPA
  k_counts<<<cdiv(E, 256), 256, 0, stream>>>(srcp, dstp, E, CNTO, CNTI);
  k_loop_scatter<<<cdiv((long long)E * 64, 256), 256, 0, stream>>>(edge_attr, dstp, E, LOOPA);
  k_edge_proj_scatter<<<dim3(cdiv(E, 16), 2), 32, 0, stream>>>(edge_attr, ep_w, ep_b,
                                                               srcp, dstp, XIN, E);
  k_norm_xinput<<<cdiv((long long)N * 256, 256), 256, 0, stream>>>(XIN, CNTO, CNTI, N);
  k_norm_loopa<<<cdiv((long long)N * 64, 256), 256, 0, stream>>>(LOOPA, CNTI, N);

  // --- GATv2 layer 1 (heads=2) ---
  k_gemm_t<256, 256><<<dim3(cdiv(N, 16), 4), 32, 0, stream>>>(XIN, g1_wl, XL, N);
  k_gemm_t<256, 256><<<dim3(cdiv(N, 16), 4), 32, 0, stream>>>(XIN, g1_wr, XR, N);
  k_init_amax<<<cdiv((long long)N * 2, 256), 256, 0, stream>>>(AMAX, N * 2);
  hipMemsetAsync(DEN, 0, (size_t)N * 2 * sizeof(float), stream);
  hipMemsetAsync(HACC, 0, (size_t)N * 256 * sizeof(float), stream);  // x_input no longer needed
  k_gat_pass1<<<Ea, 128, 0, stream>>>(XL, XR, g1_we, g1_att, edge_attr, LOOPA,
                                      srcp, dstp, E, LOG, AMAX, 2);
  k_gat_pass2<<<cdiv((long long)Ea * 2, 256), 256, 0, stream>>>(LOG, AMAX, DEN, dstp, E, Ea, 2);
  k_gat_pass3<<<cdiv((long long)Ea * 256, 256), 256, 0, stream>>>(XL, LOG, DEN, srcp, dstp,
                                                                  E, Ea, 2, HACC);
  k_gat_finalize<<<N, 128, 0, stream>>>(HACC, g1_b, n1_g, n1_b, 2, H1);

  // --- GATv2 layer 2 (heads=1) ---
  k_gemm_t<128, 128><<<dim3(cdiv(N, 16), 2), 32, 0, stream>>>(H1, g2_wl, XL, N);
  k_gemm_t<128, 128><<<dim3(cdiv(N, 16), 2), 32, 0, stream>>>(H1, g2_wr, XR, N);
  k_init_amax<<<cdiv((long long)N * 2, 256), 256, 0, stream>>>(AMAX, N * 2);
  hipMemsetAsync(DEN, 0, (size_t)N * 2 * sizeof(float), stream);
  hipMemsetAsync(HACC, 0, (size_t)N * 256 * sizeof(float), stream);
  k_gat_pass1<<<Ea, 128, 0, stream>>>(XL, XR, g2_we, g2_att, edge_attr, LOOPA,
                                      srcp, dstp, E, LOG, AMAX, 1);
  k_gat_pass2<<<cdiv((long long)Ea, 256), 256, 0, stream>>>(LOG, AMAX, DEN, dstp, E, Ea, 1);
  k_gat_pass3<<<cdiv((long long)Ea * 128, 256), 256, 0, stream>>>(XL, LOG, DEN, srcp, dstp,
                                                                  E, Ea, 1, HACC);
  k_gat_finalize<<<N, 128, 0, stream>>>(HACC, g2_b, n2_g, n2_b, 1, H2);

  // --- fused WMMA classifier -> d_out [E,1] ---
  k_classifier<<<cdiv(E, 16), 128, 0, stream>>>(H2, edge_attr, srcp, dstp,
                                                c1_w, c1_b, c2_w, c2_b, c3_w, c3_b, outp, E);
}